// LanePoolingSimple_25675314495808
// MI455X (gfx1250) — compile-verified
//
#include <hip/hip_runtime.h>
#include <hip/hip_bf16.h>
#include <stdint.h>

#define WAVE 32

// gfx1250 async global->LDS path (ASYNCcnt), guarded so compile never breaks.
#if defined(__gfx1250__) && __has_builtin(__builtin_amdgcn_global_load_async_to_lds_b64) && \
    __has_builtin(__builtin_amdgcn_s_wait_asynccnt)
#define USE_ASYNC_LDS 1
typedef int v2i __attribute__((ext_vector_type(2)));
typedef __attribute__((address_space(1))) v2i* as1_v2i_t;   // global ptr to 2xi32
typedef __attribute__((address_space(3))) v2i* as3_v2i_t;   // LDS ptr to 2xi32
#else
#define USE_ASYNC_LDS 0
#endif

// Pass 1: init per-segment packed (minbits<<32 | argmin) to all-ones.
__global__ __launch_bounds__(256) void seg_init_kernel(unsigned long long* __restrict__ segpack, int n) {
    int i = blockIdx.x * blockDim.x + threadIdx.x;
    if (i < n) segpack[i] = ~0ULL;
}

// Pass 2: segmented min+argmin. Segments are sorted -> contiguous runs, so do a
// wave32 segmented suffix-min via shuffles; only run-head lanes hit global atomics
// (scratch array is 8MB -> fully L2 resident on MI455X's 192MB L2).
// Cache policy: dist is read once -> NT load (don't cache); seg64 is re-read by
// pass 3 -> default RT so its 160MB stays resident in the 192MB L2.
__global__ __launch_bounds__(256) void seg_minpack_kernel(const float* __restrict__ dist,
                                                          const long long* __restrict__ seg64,
                                                          unsigned long long* __restrict__ segpack,
                                                          int m) {
    int i    = blockIdx.x * blockDim.x + threadIdx.x;
    int lane = threadIdx.x & (WAVE - 1);
    bool valid = (i < m);

    int seg = -1 - lane;            // unique per-lane sentinel for inactive lanes
    unsigned long long pack = ~0ULL;

#if USE_ASYNC_LDS
    // Stage the int64 segment ids through LDS with the gfx1250 async DMA path.
    __shared__ long long lds_seg[256];
    if (valid) {
        __builtin_amdgcn_global_load_async_to_lds_b64(
            (as1_v2i_t)(uintptr_t)(seg64 + i),
            (as3_v2i_t)(uint32_t)(uintptr_t)(&lds_seg[threadIdx.x]),
            0, 0);
    }
    __builtin_amdgcn_s_wait_asynccnt(0);   // own-wave async writes now visible
    if (valid) seg = (int)lds_seg[threadIdx.x];
#else
    if (valid) seg = (int)seg64[i];
#endif

    if (valid) {
        // lane_dist is uniform [0,1): non-negative floats order identically as uint bits.
        // Read-once stream -> non-temporal so it doesn't evict seg64 from L2.
        unsigned int bits = __float_as_uint(__builtin_nontemporal_load(dist + i));
        pack = ((unsigned long long)bits << 32) | (unsigned int)i;  // tie-break: smallest index
    }

    // Wave-level segmented suffix-min (runs are contiguous because seg64 is sorted).
    for (int off = 1; off < WAVE; off <<= 1) {
        unsigned long long opack = __shfl_down(pack, off, WAVE);
        int                oseg  = __shfl_down(seg,  off, WAVE);
        if ((lane + off) < WAVE && oseg == seg && opack < pack) pack = opack;
    }

    int  pseg = __shfl_up(seg, 1, WAVE);
    bool head = (lane == 0) || (pseg != seg);
    if (valid && head) atomicMin(&segpack[seg], pack);   // global_atomic_min_u64, L2-resident
}

// Pass 3: emit mask. seg64 (RT, resident from pass 2) + segpack + ts_mask hit L2;
// the output is write-once -> NT store so it streams past L2 to HBM.
__global__ __launch_bounds__(256) void seg_emit_kernel(const long long* __restrict__ seg64,
                                                       const unsigned long long* __restrict__ segpack,
                                                       const int* __restrict__ ts_mask,
                                                       float* __restrict__ out, int m) {
    int i = blockIdx.x * blockDim.x + threadIdx.x;
    if (i >= m) return;
    int seg = (int)seg64[i];
    unsigned long long pack = segpack[seg];
    unsigned int win = (unsigned int)(pack & 0xFFFFFFFFu);
    float v = (win == (unsigned int)i && ts_mask[seg] > 0) ? 1.0f : 0.0f;
    __builtin_nontemporal_store(v, out + i);
}

extern "C" void kernel_launch(void* const* d_in, const int* in_sizes, int n_in,
                              void* d_out, int out_size, void* d_ws, size_t ws_size,
                              hipStream_t stream) {
    const float*     dist  = (const float*)d_in[0];       // lane_dist: float32[M]
    const long long* seg64 = (const long long*)d_in[1];   // same_obstacle_mask: int64[M] (sorted)
    const int*       ts    = (const int*)d_in[2];         // ts_mask: int32[N]
    float*           out   = (float*)d_out;               // bool mask -> float[M]

    int m = in_sizes[0];
    int n = in_sizes[2];

    unsigned long long* segpack = (unsigned long long*)d_ws;   // needs N*8 = 8MB scratch

    const int tb = 256;
    seg_init_kernel   <<<(n + tb - 1) / tb, tb, 0, stream>>>(segpack, n);
    seg_minpack_kernel<<<(m + tb - 1) / tb, tb, 0, stream>>>(dist, seg64, segpack, m);
    seg_emit_kernel   <<<(m + tb - 1) / tb, tb, 0, stream>>>(seg64, segpack, ts, out, m);
}